// GraphAutoEncoderModel_89446988907007
// MI455X (gfx1250) — compile-verified
//
#include <hip/hip_runtime.h>
#include <stdint.h>

// Problem constants (from reference: SUPPORT=(16,16), F_DIM=64, BATCH=4096)
#define S1 16
#define S2 16
#define ROWS_PER_J 34          // 1 + S2 + 1 + S2
#define FD 64
#define FD1 65
#define ROWS_PER_B 1088        // 2 * S1 * ROWS_PER_J
#define THREADS 128

__global__ __launch_bounds__(THREADS)
void gather2hop_kernel(const int*   __restrict__ node_ids,
                       const float* __restrict__ features,
                       const int*   __restrict__ in_sample,
                       const int*   __restrict__ out_sample,
                       const float* __restrict__ in_amnt,
                       const float* __restrict__ out_amnt,
                       float* __restrict__ out_feat,
                       float* __restrict__ out_weight)
{
    __shared__ __align__(16) float s_feat[ROWS_PER_J * FD]; // 34 rows x 64 f32 = 8704 B
    __shared__ int   s_node[ROWS_PER_J];
    __shared__ float s_w[ROWS_PER_J];    // column-0 value per row
    __shared__ float s_wt[ROWS_PER_J];   // weight output per row

    const int tid    = threadIdx.x;
    const int blk    = blockIdx.x;
    const int b      = blk >> 5;     // 32 blocks per batch element
    const int rem    = blk & 31;
    const int branch = rem >> 4;     // 0 = in-branch, 1 = out-branch
    const int j      = rem & 15;     // first-hop slot

    // ---- Phase A: per-row node index + weights (34 rows) ----
    if (tid < ROWS_PER_J) {
        const int nid = node_ids[b];
        const int*   samp1 = branch ? out_sample : in_sample;
        const float* am1   = branch ? out_amnt   : in_amnt;
        const int   n1 = samp1[nid * S1 + j];
        const float w1 = am1  [nid * S1 + j];

        const int t = tid;
        int n; float w, wt;
        if (t == 0 || t == 17) {                 // first-hop row (repeated at t=17)
            n = n1; w = w1; wt = w1 * w1;
        } else if (t < 17) {                     // second hop, in-samples
            const int k = t - 1;
            n  = in_sample[n1 * S2 + k];
            w  = in_amnt  [n1 * S2 + k];
            wt = w * w1;
        } else {                                 // second hop, out-samples
            const int k = t - 18;
            n  = out_sample[n1 * S2 + k];
            w  = out_amnt  [n1 * S2 + k];
            wt = w * w1;
        }
        s_node[t] = n; s_w[t] = w; s_wt[t] = wt;
    }
    __syncthreads();

    // ---- Phase B: async gather of 34 feature rows (256 B each) into LDS ----
    // 34 rows * 16 x 16-byte segments = 544 async b128 copies, direct to LDS.
    for (int seg = tid; seg < ROWS_PER_J * 16; seg += THREADS) {
        const int row  = seg >> 4;
        const int part = seg & 15;
        const float* gsrc = features + (size_t)s_node[row] * FD + part * 4;
        // Generic LDS pointer low 32 bits == LDS byte offset (ISA 10.2 aperture rule).
        unsigned lds = (unsigned)(unsigned long long)(const void*)(&s_feat[row * FD + part * 4]);
        unsigned long long ga = (unsigned long long)gsrc;
        asm volatile("global_load_async_to_lds_b128 %0, %1, off"
                     :: "v"(lds), "v"(ga)
                     : "memory");
    }
    asm volatile("s_wait_asynccnt 0" ::: "memory");
    __syncthreads();

    // ---- Phase C: coalesced non-temporal stores of 34 x 65 output floats ----
    const size_t row_base = (size_t)b * ROWS_PER_B
                          + (size_t)branch * (S1 * ROWS_PER_J)
                          + (size_t)j * ROWS_PER_J;
    for (int e = tid; e < ROWS_PER_J * FD1; e += THREADS) {
        const int t = e / FD1;
        const int c = e - t * FD1;
        const float v = (c == 0) ? s_w[t] : s_feat[t * FD + (c - 1)];
        __builtin_nontemporal_store(v, &out_feat[(row_base + t) * FD1 + c]);
    }
    if (tid < ROWS_PER_J) {
        __builtin_nontemporal_store(s_wt[tid], &out_weight[row_base + tid]);
    }
}

extern "C" void kernel_launch(void* const* d_in, const int* in_sizes, int n_in,
                              void* d_out, int out_size, void* d_ws, size_t ws_size,
                              hipStream_t stream)
{
    const int*   node_ids   = (const int*)  d_in[0];
    const float* features   = (const float*)d_in[1];
    const int*   in_sample  = (const int*)  d_in[2];
    const int*   out_sample = (const int*)  d_in[3];
    const float* in_amnt    = (const float*)d_in[4];
    const float* out_amnt   = (const float*)d_in[5];

    const int B = in_sizes[0];                       // 4096
    float* out_feat   = (float*)d_out;
    float* out_weight = out_feat + (size_t)B * ROWS_PER_B * FD1;

    dim3 grid((unsigned)B * 32);                     // one block per (b, branch, j)
    gather2hop_kernel<<<grid, THREADS, 0, stream>>>(
        node_ids, features, in_sample, out_sample, in_amnt, out_amnt,
        out_feat, out_weight);
}